// GCNNet_59983513256468
// MI455X (gfx1250) — compile-verified
//
#include <hip/hip_runtime.h>

#define D 320
#define D4 80            // D / 4
#define N_GRAPHS 64

typedef __attribute__((ext_vector_type(2))) float v2f;
typedef __attribute__((ext_vector_type(8))) float v8f;

__device__ __forceinline__ void atomAddF32(float* p, float v) {
  __hip_atomic_fetch_add(p, v, __ATOMIC_RELAXED, __HIP_MEMORY_SCOPE_AGENT);
}

// ---------------- WMMA f32 GEMM: C[M,N] = A[M,K] * B[K,N] ----------------
// One wave -> 16x64 strip of C. 4 waves per block take consecutive M-tiles.
// Requires M%16==0, K%4==0, N%64==0 (holds: 50000, 320, 320).
__global__ __launch_bounds__(128) void gemm_wmma_f32(
    const float* __restrict__ A, const float* __restrict__ B,
    float* __restrict__ C, int M, int K, int N) {
  const int lane  = threadIdx.x & 31;
  const int wave  = threadIdx.x >> 5;
  const int mtile = blockIdx.x * 4 + wave;
  if (mtile * 16 >= M) return;              // wave-uniform exit (EXEC stays all-1s)
  const int n0  = blockIdx.y * 64;
  const int row = lane & 15;                // M index for A, N index for B/C
  const int kh  = lane >> 4;                // K-half select per ISA layout

  v8f acc0 = {}, acc1 = {}, acc2 = {}, acc3 = {};

  const float* Ap = A + (size_t)(mtile * 16 + row) * K + 2 * kh;
  const float* Bp = B + (size_t)(2 * kh) * N + n0 + row;

  for (int k0 = 0; k0 < K; k0 += 4) {
    // A 16x4 fragment: lanes 0-15 K(0,1), lanes 16-31 K(2,3) -> contiguous float2
    float2 av = *(const float2*)(Ap + k0);
    v2f a; a.x = av.x; a.y = av.y;
    // B 4x16 fragments: lanes 0-15 rows K(0,1), lanes 16-31 rows K(2,3)
    const float* b0p = Bp + (size_t)k0 * N;
    const float* b1p = b0p + N;
    v2f b0; b0.x = b0p[ 0]; b0.y = b1p[ 0];
    v2f b1; b1.x = b0p[16]; b1.y = b1p[16];
    v2f b2; b2.x = b0p[32]; b2.y = b1p[32];
    v2f b3; b3.x = b0p[48]; b3.y = b1p[48];
    acc0 = __builtin_amdgcn_wmma_f32_16x16x4_f32(false, a, false, b0, (short)0, acc0, false, false);
    acc1 = __builtin_amdgcn_wmma_f32_16x16x4_f32(false, a, false, b1, (short)0, acc1, false, false);
    acc2 = __builtin_amdgcn_wmma_f32_16x16x4_f32(false, a, false, b2, (short)0, acc2, false, false);
    acc3 = __builtin_amdgcn_wmma_f32_16x16x4_f32(false, a, false, b3, (short)0, acc3, false, false);
  }

  // C/D layout: VGPR i, lanes 0-15 -> M=i ; lanes 16-31 -> M=i+8 ; lane%16 = N
  float* Cp = C + (size_t)(mtile * 16 + 8 * kh) * N + n0 + row;
#pragma unroll
  for (int i = 0; i < 8; ++i) {
    Cp[(size_t)i * N +  0] = acc0[i];
    Cp[(size_t)i * N + 16] = acc1[i];
    Cp[(size_t)i * N + 32] = acc2[i];
    Cp[(size_t)i * N + 48] = acc3[i];
  }
}

// ---------------- graph helper kernels ----------------
__global__ void fill_f32(float* __restrict__ p, float v, int n) {
  int i = blockIdx.x * blockDim.x + threadIdx.x;
  if (i < n) p[i] = v;
}

__global__ void deg_edges(const int* __restrict__ dst, float* __restrict__ deg, int E) {
  int e = blockIdx.x * blockDim.x + threadIdx.x;
  if (e < E) atomAddF32(&deg[dst[e]], 1.0f);
}

__global__ void rsqrt_k(float* __restrict__ d, int n) {
  int i = blockIdx.x * blockDim.x + threadIdx.x;
  if (i < n) d[i] = rsqrtf(d[i]);
}

// agg[i][:] = h[i][:]*dinv[i]^2 + b[:]   (self-loop term + bias), float4 lanes
__global__ void init_agg_v4(const float4* __restrict__ h4, const float* __restrict__ dinv,
                            const float4* __restrict__ b4, float4* __restrict__ agg4,
                            int total) {
  int id = blockIdx.x * blockDim.x + threadIdx.x;   // over n*D4
  if (id >= total) return;
  int i = id / D4, d = id % D4;
  float di = dinv[i];
  float s  = di * di;
  float4 hv = h4[id];
  float4 bv = b4[d];
  float4 o;
  o.x = hv.x * s + bv.x;
  o.y = hv.y * s + bv.y;
  o.z = hv.z * s + bv.z;
  o.w = hv.w * s + bv.w;
  agg4[id] = o;
}

// 2 edges per 160-thread block; 80 threads/edge, float4 gather + 4 atomics each.
__global__ __launch_bounds__(160) void edge_agg_v4(
    const float4* __restrict__ h4, const float* __restrict__ dinv,
    const int* __restrict__ src, const int* __restrict__ dst,
    float* __restrict__ agg, int E) {
  int tid = threadIdx.x;
  int e   = blockIdx.x * 2 + tid / D4;
  if (e >= E) return;
  int d = tid % D4;
  int s = src[e], t = dst[e];
  float c = dinv[s] * dinv[t];
  float4 hv = h4[(size_t)s * D4 + d];
  float* p = agg + (size_t)t * D + 4 * d;
  atomAddF32(p + 0, hv.x * c);
  atomAddF32(p + 1, hv.y * c);
  atomAddF32(p + 2, hv.z * c);
  atomAddF32(p + 3, hv.w * c);
}

__global__ void relu_v4(float4* __restrict__ p, int n4) {
  int i = blockIdx.x * blockDim.x + threadIdx.x;
  if (i < n4) {
    float4 v = p[i];
    v.x = fmaxf(v.x, 0.0f);
    v.y = fmaxf(v.y, 0.0f);
    v.z = fmaxf(v.z, 0.0f);
    v.w = fmaxf(v.w, 0.0f);
    p[i] = v;
  }
}

// segment-max pool; inputs are >=0 (post-ReLU) so uint atomicMax == float max
__global__ void pool_max_v4(const float4* __restrict__ h4, const int* __restrict__ batch,
                            float* __restrict__ g, int total) {
  int id = blockIdx.x * blockDim.x + threadIdx.x;   // over n*D4
  if (id >= total) return;
  int i = id / D4, d = id % D4;
  float4 v = h4[id];
  unsigned* gp = (unsigned*)(g + (size_t)batch[i] * D + 4 * d);
  atomicMax(gp + 0, __float_as_uint(v.x));
  atomicMax(gp + 1, __float_as_uint(v.y));
  atomicMax(gp + 2, __float_as_uint(v.z));
  atomicMax(gp + 3, __float_as_uint(v.w));
}

// tiny dense layer: Y[rows,Ncols] = act(X[rows,K] @ W[K,Ncols] + b)
__global__ void mlp(const float* __restrict__ X, const float* __restrict__ W,
                    const float* __restrict__ b, float* __restrict__ Y,
                    int K, int Ncols, int rows, int doRelu) {
  int id = blockIdx.x * blockDim.x + threadIdx.x;
  if (id >= rows * Ncols) return;
  int r = id / Ncols, c = id % Ncols;
  float s = b[c];
  for (int k = 0; k < K; ++k) s += X[(size_t)r * K + k] * W[(size_t)k * Ncols + c];
  Y[id] = doRelu ? fmaxf(s, 0.0f) : s;
}

extern "C" void kernel_launch(void* const* d_in, const int* in_sizes, int n_in,
                              void* d_out, int out_size, void* d_ws, size_t ws_size,
                              hipStream_t stream) {
  const float* x   = (const float*)d_in[0];
  const float* W1  = (const float*)d_in[1];  const float* b1  = (const float*)d_in[2];
  const float* W2  = (const float*)d_in[3];  const float* b2  = (const float*)d_in[4];
  const float* W3  = (const float*)d_in[5];  const float* b3  = (const float*)d_in[6];
  const float* Wf1 = (const float*)d_in[7];  const float* bf1 = (const float*)d_in[8];
  const float* Wf2 = (const float*)d_in[9];  const float* bf2 = (const float*)d_in[10];
  const float* Wf3 = (const float*)d_in[11]; const float* bf3 = (const float*)d_in[12];
  const int* edge  = (const int*)d_in[13];
  const int* batch = (const int*)d_in[14];

  const int E = in_sizes[13] / 2;       // 800000
  const int n = in_sizes[0] / D;        // 50000
  const int* src = edge;
  const int* dst = edge + E;

  // workspace carve-out (~128.3 MB of f32)
  float* h    = (float*)d_ws;                 // [n, D]
  float* agg  = h    + (size_t)n * D;         // [n, D]
  float* dinv = agg  + (size_t)n * D;         // [n]
  float* g    = dinv + n;                     // [64, 320]
  float* g1   = g    + N_GRAPHS * D;          // [64, 256]
  float* g2   = g1   + N_GRAPHS * 256;        // [64, 16]

  const int nd4 = n * D4;               // n*D/4 vector elements

  // degrees: deg = 1 (self loop) + in-degree(dst); dinv = rsqrt(deg)
  fill_f32<<<(n + 255) / 256, 256, 0, stream>>>(dinv, 1.0f, n);
  deg_edges<<<(E + 255) / 256, 256, 0, stream>>>(dst, dinv, E);
  rsqrt_k<<<(n + 255) / 256, 256, 0, stream>>>(dinv, n);

  const float* Ws[3] = {W1, W2, W3};
  const float* bs[3] = {b1, b2, b3};
  const float* layer_in = x;
  dim3 ggrid((n / 16 + 3) / 4, D / 64);   // 782 x 5

  for (int l = 0; l < 3; ++l) {
    gemm_wmma_f32<<<ggrid, 128, 0, stream>>>(layer_in, Ws[l], h, n, D, D);
    init_agg_v4<<<(nd4 + 255) / 256, 256, 0, stream>>>(
        (const float4*)h, dinv, (const float4*)bs[l], (float4*)agg, nd4);
    edge_agg_v4<<<(E + 1) / 2, 160, 0, stream>>>(
        (const float4*)h, dinv, src, dst, agg, E);
    relu_v4<<<(nd4 + 255) / 256, 256, 0, stream>>>((float4*)agg, nd4);
    layer_in = agg;
  }

  // global max pool + MLP head
  fill_f32<<<(N_GRAPHS * D + 255) / 256, 256, 0, stream>>>(g, 0.0f, N_GRAPHS * D);
  pool_max_v4<<<(nd4 + 255) / 256, 256, 0, stream>>>((const float4*)agg, batch, g, nd4);
  mlp<<<(N_GRAPHS * 256 + 255) / 256, 256, 0, stream>>>(g,  Wf1, bf1, g1, D,   256, N_GRAPHS, 1);
  mlp<<<(N_GRAPHS * 16  + 255) / 256, 256, 0, stream>>>(g1, Wf2, bf2, g2, 256, 16,  N_GRAPHS, 1);
  mlp<<<1, N_GRAPHS, 0, stream>>>(g2, Wf3, bf3, (float*)d_out, 16, 1, N_GRAPHS, 0);
}